// CPUBouncingEmbedding_79946521247869
// MI455X (gfx1250) — compile-verified
//
#include <hip/hip_runtime.h>
#include <hip/hip_bf16.h>
#include <stdint.h>

// CPUBouncingEmbedding for MI455X (gfx1250).
// Fused: stochastic-round(fp32->bf16) + embedding gather + *32 scale.
// Memory-bound: ~64MB read + 32MB write => ~4.3us @ 23.3 TB/s (197MB fp32
// table ~fits the 192MB L2, so repeated vocab rows hit L2).
// Data path: per-wave async global->LDS ring, depth 4, 3 rows in flight
// (global_load_async_to_lds_b128 / s_wait_asynccnt). Each wave owns a
// disjoint 128-column slice of every row -> zero block barriers.

#define EMB_DIM        1024
#define TPB_TOKENS     16         // tokens per block
#define THREADS        256        // 8 wave32s
#define WAVES          8
#define COLS_PER_WAVE  (EMB_DIM / WAVES)   // 128
#define COLS_PER_LANE  4                   // one b128 load / lane
#define RING           4                   // LDS row buffers (16KB/block)

#if defined(__gfx1250__) && \
    __has_builtin(__builtin_amdgcn_global_load_async_to_lds_b128) && \
    __has_builtin(__builtin_amdgcn_s_wait_asynccnt)
#define USE_ASYNC_LDS 1
#else
#define USE_ASYNC_LDS 0
#endif

// Builtin signature (from hipcc diagnostic): non-const AS1 int4* source,
// AS3 int4* LDS destination, imm offset, imm cpol.
typedef int v4i __attribute__((ext_vector_type(4)));
typedef __attribute__((address_space(1))) v4i* gv4_ptr_t;
typedef __attribute__((address_space(3))) v4i* lv4_ptr_t;

// Deterministic per-table-entry hash -> 16-bit stochastic-rounding addend.
// Pure function of (row*DIM + col): same vocab row always rounds identically,
// matching the reference's "round the table once, then gather" semantics.
__device__ __forceinline__ uint32_t mix32(uint32_t x) {
    x ^= x >> 16; x *= 0x7FEB352Du;
    x ^= x >> 15; x *= 0x846CA68Bu;
    x ^= x >> 16;
    return x;
}

// Stochastic round fp32 -> bf16 bits, then *32 (exact power-of-2 in bf16).
__device__ __forceinline__ uint32_t sr_scale_bf16(float w, uint32_t h) {
    uint32_t bits = __float_as_uint(w);
    bits = (bits + (h & 0xFFFFu)) & 0xFFFF0000u;       // SR truncate to bf16
    float f = __uint_as_float(bits) * 32.0f;           // exponent shift, exact
    return __float_as_uint(f) >> 16;                   // bf16 bit pattern
}

__device__ __forceinline__ void convert_store(float4 v, uint32_t row, int col,
                                              uint32_t* __restrict__ out,
                                              size_t tok) {
    const uint32_t base = row * (uint32_t)EMB_DIM + (uint32_t)col;
    uint32_t b0 = sr_scale_bf16(v.x, mix32(base + 0u));
    uint32_t b1 = sr_scale_bf16(v.y, mix32(base + 1u));
    uint32_t b2 = sr_scale_bf16(v.z, mix32(base + 2u));
    uint32_t b3 = sr_scale_bf16(v.w, mix32(base + 3u));
    uint2 pk;
    pk.x = b0 | (b1 << 16);
    pk.y = b2 | (b3 << 16);
    // 8B packed store: out is bf16 viewed as uint32 pairs
    *reinterpret_cast<uint2*>(out + tok * (EMB_DIM / 2) + (col >> 1)) = pk;
}

extern "C" __global__ void __launch_bounds__(THREADS)
embed_sr_gather(const int* __restrict__ ids,
                const float* __restrict__ weight,
                uint32_t* __restrict__ out) {
    const int wave = (int)(threadIdx.x >> 5);   // wave32
    const int lane = (int)(threadIdx.x & 31);
    const int col  = wave * COLS_PER_WAVE + lane * COLS_PER_LANE;
    const size_t t0 = (size_t)blockIdx.x * TPB_TOKENS;

#if USE_ASYNC_LDS
    // Ring of 4 row buffers; each wave touches only its 128-col slice.
    __shared__ float lds[RING][EMB_DIM];

    auto issue = [&](int tok_i, int buf) {
        const uint32_t r = (uint32_t)ids[t0 + tok_i];   // uniform -> s_load
        const float* s = weight + (size_t)r * EMB_DIM + col;
        __builtin_amdgcn_global_load_async_to_lds_b128(
            (gv4_ptr_t)s, (lv4_ptr_t)&lds[buf][col], 0, 0);
    };

    // Prologue: 2 rows in flight before the loop; iteration i issues row i+2,
    // keeping 3 outstanding. wait asynccnt<=2 then retires exactly row i
    // (async loads complete in order within a wave).
    issue(0, 0);
    issue(1, 1);

    #pragma unroll
    for (int i = 0; i < TPB_TOKENS; ++i) {
        const uint32_t row = (uint32_t)ids[t0 + i];
        if (i + 2 < TPB_TOKENS) {
            issue(i + 2, (i + 2) & (RING - 1));
            __builtin_amdgcn_s_wait_asynccnt(2);   // row i ready, i+1/i+2 in flight
        } else if (i + 1 < TPB_TOKENS) {
            __builtin_amdgcn_s_wait_asynccnt(1);   // row i ready, i+1 in flight
        } else {
            __builtin_amdgcn_s_wait_asynccnt(0);   // drain
        }
        __asm__ volatile("" ::: "memory");         // LDS written behind compiler's back
        float4 v = *reinterpret_cast<const float4*>(&lds[i & (RING - 1)][col]);
        convert_store(v, row, col, out, t0 + i);
    }
#else
    // Fallback: register-resident 2-deep software pipeline of b128 loads.
    uint32_t row = (uint32_t)ids[t0];
    float4 v = *reinterpret_cast<const float4*>(weight + (size_t)row * EMB_DIM + col);
    #pragma unroll
    for (int i = 0; i < TPB_TOKENS; ++i) {
        uint32_t row_cur = row;
        float4 v_cur = v;
        if (i + 1 < TPB_TOKENS) {
            row = (uint32_t)ids[t0 + i + 1];
            v = *reinterpret_cast<const float4*>(weight + (size_t)row * EMB_DIM + col);
        }
        convert_store(v_cur, row_cur, col, out, t0 + i);
    }
#endif
}

// Tail: one block per leftover token (not hit for 16384 tokens, kept for safety).
extern "C" __global__ void __launch_bounds__(THREADS)
embed_sr_tail(const int* __restrict__ ids,
              const float* __restrict__ weight,
              uint32_t* __restrict__ out,
              int first_tok) {
    const size_t tok = (size_t)first_tok + blockIdx.x;
    const int col = (int)threadIdx.x * COLS_PER_LANE;
    const uint32_t row = (uint32_t)ids[tok];
    float4 v = *reinterpret_cast<const float4*>(weight + (size_t)row * EMB_DIM + col);
    convert_store(v, row, col, out, tok);
}

extern "C" void kernel_launch(void* const* d_in, const int* in_sizes, int n_in,
                              void* d_out, int out_size, void* d_ws, size_t ws_size,
                              hipStream_t stream) {
    const int*   ids    = (const int*)d_in[0];     // [B*S] int32
    const float* weight = (const float*)d_in[1];   // [VOCAB, DIM] fp32
    uint32_t*    out    = (uint32_t*)d_out;        // [B*S, DIM] bf16 (packed pairs)

    const int n_tokens = in_sizes[0];
    const int nblk = n_tokens / TPB_TOKENS;
    const int rem  = n_tokens % TPB_TOKENS;

    if (nblk > 0)
        embed_sr_gather<<<nblk, THREADS, 0, stream>>>(ids, weight, out);
    if (rem > 0)
        embed_sr_tail<<<rem, THREADS, 0, stream>>>(ids, weight, out, n_tokens - rem);
}